// BiAttention_34256659153649
// MI455X (gfx1250) — compile-verified
//
#include <hip/hip_runtime.h>
#include <hip/hip_bf16.h>
#include <math.h>

typedef __attribute__((ext_vector_type(2))) float v2f;
typedef __attribute__((ext_vector_type(8))) float v8f;

#define HID 1024
#define BSZ 8
#define LD  2048
#define LM  512

#if defined(__gfx1250__) && __has_builtin(__builtin_amdgcn_wmma_f32_16x16x4_f32)
#define HAVE_WMMA_F32 1
#else
#define HAVE_WMMA_F32 0
#endif

__device__ inline float wave_sum(float v) {
#pragma unroll
  for (int off = 16; off > 0; off >>= 1) v += __shfl_xor(v, off, 32);
  return v;
}

// Non-temporal store: keep the 256MB output stream out of L2 so that
// input (64MB) and W_in2 (4MB) stay resident across their many rereads.
__device__ inline void store_nt(float* p, float v) {
  __builtin_nontemporal_store(v, p);
}

// ---- A1: per-row dot with a weight vector (input_dot / memory_dot) --------
__global__ void __launch_bounds__(256)
dot_rows_kernel(const float* __restrict__ src, const float* __restrict__ w,
                float* __restrict__ dst, int nrows) {
  int wid = threadIdx.x >> 5, lane = threadIdx.x & 31;
  int row = blockIdx.x * 8 + wid;
  if (row >= nrows) return;
  const float* p = src + (size_t)row * HID;
  float s = 0.f;
  for (int k = lane; k < HID; k += 32) s += p[k] * w[k];
  s = wave_sum(s);
  if (lane == 0) dst[row] = s;
}

// ---- A2: softmaxes. blockIdx.x==0 -> wone over LM (with mask bias),
//                     blockIdx.x==1 -> wtwo over LD -------------------------
__global__ void __launch_bounds__(256)
softmax_kernel(const float* __restrict__ input_dot,
               const float* __restrict__ memory_dot,
               const float* __restrict__ mask,
               float* __restrict__ wone, float* __restrict__ wtwo) {
  int b = blockIdx.y;
  int which = blockIdx.x;  // 0 -> wone, 1 -> wtwo
  int n = which ? LD : LM;
  const float* base = which ? (input_dot + b * LD) : (memory_dot + b * LM);
  const float* mk = mask + b * LM;
  float* dst = which ? (wtwo + b * LD) : (wone + b * LM);
  __shared__ float red[256];
  int tid = threadIdx.x;
  auto score = [&](int i) -> float {
    float s = base[i];
    if (!which) s -= 1e30f * (1.0f - mk[i]);
    return s;
  };
  float mx = -3.0e38f;
  for (int i = tid; i < n; i += 256) mx = fmaxf(mx, score(i));
  red[tid] = mx; __syncthreads();
  for (int s2 = 128; s2 > 0; s2 >>= 1) {
    if (tid < s2) red[tid] = fmaxf(red[tid], red[tid + s2]);
    __syncthreads();
  }
  mx = red[0]; __syncthreads();
  float sm = 0.f;
  for (int i = tid; i < n; i += 256) sm += __expf(score(i) - mx);
  red[tid] = sm; __syncthreads();
  for (int s2 = 128; s2 > 0; s2 >>= 1) {
    if (tid < s2) red[tid] += red[tid + s2];
    __syncthreads();
  }
  float inv = 1.0f / red[0];
  for (int i = tid; i < n; i += 256) dst[i] = __expf(score(i) - mx) * inv;
}

// ---- A3: rmem[b,k] = sum_m wone[b,m]*memory[b,m,k];
//          rin[b,k]  = sum_l wtwo[b,l]*input[b,l,k] -------------------------
__global__ void __launch_bounds__(256)
weighted_sum_kernel(const float* __restrict__ input,
                    const float* __restrict__ memory,
                    const float* __restrict__ wone, const float* __restrict__ wtwo,
                    float* __restrict__ rin, float* __restrict__ rmem) {
  int k = blockIdx.x * 256 + threadIdx.x;  // 0..HID-1 (coalesced)
  int b = blockIdx.y;
  if (blockIdx.z == 0) {
    const float* w = wone + b * LM;
    const float* p = memory + (size_t)b * LM * HID + k;
    float acc = 0.f;
    for (int m = 0; m < LM; ++m) acc += w[m] * p[(size_t)m * HID];
    rmem[b * HID + k] = acc;
  } else {
    const float* w = wtwo + b * LD;
    const float* p = input + (size_t)b * LD * HID + k;
    float acc = 0.f;
    for (int l = 0; l < LD; ++l) acc += w[l] * p[(size_t)l * HID];
    rin[b * HID + k] = acc;
  }
}

// ---- B: ctx = rmem @ W_mem2^T + b_mem2 ; otwo = rin @ W_in2^T + b_in2 -----
__global__ void __launch_bounds__(256)
small_gemm_kernel(const float* __restrict__ rmem, const float* __restrict__ rin,
                  const float* __restrict__ W_in2, const float* __restrict__ b_in2,
                  const float* __restrict__ W_mem2, const float* __restrict__ b_mem2,
                  float* __restrict__ ctx, float* __restrict__ otwo) {
  int wid = threadIdx.x >> 5, lane = threadIdx.x & 31;
  int w = blockIdx.x * 8 + wid;  // 0..16383
  int z = w >> 13;               // 0: ctx, 1: otwo
  int b = (w >> 10) & 7;
  int d = w & (HID - 1);
  const float* vec  = z ? (rin + b * HID) : (rmem + b * HID);
  const float* Wrow = (z ? W_in2 : W_mem2) + (size_t)d * HID;
  float s = 0.f;
  for (int k = lane; k < HID; k += 32) s += vec[k] * Wrow[k];
  s = wave_sum(s);
  if (lane == 0) {
    if (z) otwo[b * HID + d] = s + b_in2[d];
    else   ctx[b * HID + d]  = s + b_mem2[d];
  }
}

// ---- C: inp2 = input @ W_in2^T + b  (WMMA f32 16x16x4), fused epilogue ----
// Each wave: one 16(M)x64(N) tile over K=1024. Writes out block0 (inp2)
// and block2 (inp2 * ctx broadcast) with non-temporal stores.
__global__ void __launch_bounds__(256)
gemm_epilogue_kernel(const float* __restrict__ input, const float* __restrict__ W,
                     const float* __restrict__ bias, const float* __restrict__ ctx,
                     float* __restrict__ out) {
  const int K = HID;
  int wid = threadIdx.x >> 5, lane = threadIdx.x & 31;
  int w = blockIdx.x * 8 + wid;
  int mTile = w >> 4;        // 0..1023
  int nGroup = w & 15;       // 0..15
  int m0 = mTile * 16;
  int j0 = nGroup * 64;
  int lm = lane & 15;
  int half = lane >> 4;

  v8f accs[4] = {};
  const float* aPtr = input + (size_t)(m0 + lm) * K + half * 2;
  const float* bPtr = W + (size_t)(j0 + lm) * K + half * 2;
#if HAVE_WMMA_F32
  for (int k = 0; k < K; k += 4) {
    v2f a  = *(const v2f*)(aPtr + k);
    v2f b0 = *(const v2f*)(bPtr + k);
    v2f b1 = *(const v2f*)(bPtr + (size_t)16 * K + k);
    v2f b2 = *(const v2f*)(bPtr + (size_t)32 * K + k);
    v2f b3 = *(const v2f*)(bPtr + (size_t)48 * K + k);
    accs[0] = __builtin_amdgcn_wmma_f32_16x16x4_f32(false, a, false, b0, (short)0, accs[0], false, false);
    accs[1] = __builtin_amdgcn_wmma_f32_16x16x4_f32(false, a, false, b1, (short)0, accs[1], false, false);
    accs[2] = __builtin_amdgcn_wmma_f32_16x16x4_f32(false, a, false, b2, (short)0, accs[2], false, false);
    accs[3] = __builtin_amdgcn_wmma_f32_16x16x4_f32(false, a, false, b3, (short)0, accs[3], false, false);
  }
#else
  for (int t = 0; t < 4; ++t) {
    int col = j0 + t * 16 + lm;
    for (int r = 0; r < 8; ++r) {
      int row = m0 + r + half * 8;
      float s = 0.f;
      for (int k = 0; k < K; ++k) s += input[(size_t)row * K + k] * W[(size_t)col * K + k];
      accs[t][r] = s;
    }
  }
#endif
  // Whole tile lies within one batch (m0 multiple of 16, LD multiple of 16).
  int bb = m0 >> 11;          // row / LD
  int l0 = m0 & (LD - 1);
  size_t outBase = (size_t)bb * (4 * LD) * HID;
#pragma unroll
  for (int t = 0; t < 4; ++t) {
    int col = j0 + t * 16 + lm;
    float bv = bias[col];
    float cv = ctx[bb * HID + col];
#pragma unroll
    for (int r = 0; r < 8; ++r) {
      int l = l0 + r + half * 8;   // C layout: VGPR r -> row r (lanes<16), 8+r (lanes>=16)
      float v = accs[t][r] + bv;
      size_t o0 = outBase + (size_t)l * HID + col;
      store_nt(out + o0, v);                                  // block0: inp2
      store_nt(out + o0 + (size_t)(2 * LD) * HID, v * cv);    // block2: inp2*output_one
    }
  }
}

// ---- D: broadcast rows: block1 = ctx[b,:], block3 = otwo[b,:]*ctx[b,:] ----
__global__ void __launch_bounds__(256)
broadcast_kernel(const float* __restrict__ ctx, const float* __restrict__ otwo,
                 float* __restrict__ out) {
  size_t idx = (size_t)blockIdx.x * 256 + threadIdx.x;  // BSZ*LD*HID threads
  int d = (int)(idx & (HID - 1));
  int l = (int)((idx >> 10) & (LD - 1));
  int b = (int)(idx >> 21);
  float c = ctx[b * HID + d];
  size_t base = (size_t)b * (4 * LD) * HID;
  store_nt(out + base + (size_t)(LD + l) * HID + d, c);                        // output_one
  store_nt(out + base + (size_t)(3 * LD + l) * HID + d, otwo[b * HID + d] * c); // output_two*output_one
}

extern "C" void kernel_launch(void* const* d_in, const int* in_sizes, int n_in,
                              void* d_out, int out_size, void* d_ws, size_t ws_size,
                              hipStream_t stream) {
  const float* input  = (const float*)d_in[0];
  const float* memory = (const float*)d_in[1];
  const float* mask   = (const float*)d_in[2];
  const float* w_in1  = (const float*)d_in[3];
  const float* w_mem1 = (const float*)d_in[4];
  const float* W_in2  = (const float*)d_in[5];
  const float* b_in2  = (const float*)d_in[6];
  const float* W_mem2 = (const float*)d_in[7];
  const float* b_mem2 = (const float*)d_in[8];
  float* out = (float*)d_out;

  float* ws = (float*)d_ws;                 // ~295 KB of f32 scratch
  float* input_dot  = ws;                   // 16384
  float* memory_dot = ws + 16384;           // 4096
  float* wone       = ws + 20480;           // 4096
  float* wtwo       = ws + 24576;           // 16384
  float* rmem       = ws + 40960;           // 8192
  float* rin        = ws + 49152;           // 8192
  float* ctx        = ws + 57344;           // 8192
  float* otwo       = ws + 65536;           // 8192

  dot_rows_kernel<<<BSZ * LD / 8, 256, 0, stream>>>(input, w_in1, input_dot, BSZ * LD);
  dot_rows_kernel<<<BSZ * LM / 8, 256, 0, stream>>>(memory, w_mem1, memory_dot, BSZ * LM);
  softmax_kernel<<<dim3(2, BSZ), 256, 0, stream>>>(input_dot, memory_dot, mask, wone, wtwo);
  weighted_sum_kernel<<<dim3(HID / 256, BSZ, 2), 256, 0, stream>>>(input, memory, wone, wtwo, rin, rmem);
  small_gemm_kernel<<<16384 / 8, 256, 0, stream>>>(rmem, rin, W_in2, b_in2, W_mem2, b_mem2, ctx, otwo);
  gemm_epilogue_kernel<<<(1024 * 16) / 8, 256, 0, stream>>>(input, W_in2, b_in2, ctx, out);
  broadcast_kernel<<<(size_t)(BSZ) * LD * HID / 256, 256, 0, stream>>>(ctx, otwo, out);
}